// TinyGRU_6004364279976
// MI455X (gfx1250) — compile-verified
//
#include <hip/hip_runtime.h>

// ---------------------------------------------------------------------------
// TinyGRU for MI455X (gfx1250, wave32)
//   B=4096, S=2048, I=3, H=4, O=2, 3H=12
// Pass 1: x_gates = inputs @ W_ih^T + b_ih  via V_WMMA_F32_16X16X4_F32
//         (K padded 3->4 with 1.0; bias folded into B-matrix row K=3)
// Pass 2: sequential GRU recurrence, one quad (4 lanes) per batch element,
//         h broadcast inside the quad via DPP quad_perm; readout fused.
// S is chunked so the x_gates slab stays L2-resident (<=50MB per chunk).
// ---------------------------------------------------------------------------

typedef __attribute__((ext_vector_type(2))) float v2f;
typedef __attribute__((ext_vector_type(8))) float v8f;

#define NB 4096
#define NS 2048
#define NI 3
#define NH 4
#define NO 2
#define NG 12  // 3*H

// ---- fast transcendentals (hardware trans ops) ----------------------------
__device__ __forceinline__ float fexp2(float x) {
#if __has_builtin(__builtin_amdgcn_exp2f)
  return __builtin_amdgcn_exp2f(x);
#else
  return exp2f(x);
#endif
}
__device__ __forceinline__ float frcp(float x) {
#if __has_builtin(__builtin_amdgcn_rcpf)
  return __builtin_amdgcn_rcpf(x);
#else
  return 1.0f / x;
#endif
}
__device__ __forceinline__ float fsigmoid(float x) {
  // 1/(1+exp(-x)) = rcp(1 + 2^(-x*log2e))
  return frcp(1.0f + fexp2(-1.4426950408889634f * x));
}
__device__ __forceinline__ float ftanh_fast(float x) {
#if __has_builtin(__builtin_amdgcn_tanhf)
  return __builtin_amdgcn_tanhf(x);
#else
  // tanh(x) = 1 - 2/(2^(2x*log2e)+1)
  return __builtin_fmaf(-2.0f, frcp(fexp2(2.8853900817779268f * x) + 1.0f), 1.0f);
#endif
}

// ---- quad broadcast: every lane gets quad-lane Q's value ------------------
template <int Q>
__device__ __forceinline__ float qbcast(float x) {
#if __has_builtin(__builtin_amdgcn_update_dpp)
  // DPP quad_perm [Q,Q,Q,Q] -> ctrl = Q*0x55 ; row_mask/bank_mask all on
  return __int_as_float(
      __builtin_amdgcn_update_dpp(0, __float_as_int(x), Q * 0x55, 0xF, 0xF, true));
#else
  // ds_swizzle group-of-4 mode: offset[15]=1, 2-bit sel per quad lane
  return __int_as_float(
      __builtin_amdgcn_ds_swizzle(__float_as_int(x), 0x8000 | (Q * 0x55)));
#endif
}

// ---------------------------------------------------------------------------
// Pass 1: WMMA input projection for one S-chunk.
// Chunk rows r in [0, NB*sc): b = r/sc, s = s0 + (r - b*sc).
// xg layout: [r][12] (48B rows, contiguous per (b, s-chunk) stream).
// One wave handles 16 rows via V_WMMA_F32_16X16X4_F32.
// ---------------------------------------------------------------------------
__global__ __launch_bounds__(256) void tinygru_xgates_wmma(
    const float* __restrict__ in,    // [B,S,3]
    const float* __restrict__ W_ih,  // [12,3]
    const float* __restrict__ b_ih,  // [12]
    float* __restrict__ xg,          // [NB*sc, 12] chunk slab
    int s0, int sc) {
  const int wave = (blockIdx.x << 3) | (threadIdx.x >> 5);
  const int lane = threadIdx.x & 31;
  const int n = lane & 15;   // A-row index / B,D column index
  const int kh = lane >> 4;  // K half: 0 -> K0,K1 ; 1 -> K2,K3

  // B fragment: B[k][n] = W_ih[n][k] (k<3), B[3][n] = b_ih[n]; cols>=12 zero.
  v2f bm;
  if (n < NG) {
    if (kh == 0) {
      bm.x = W_ih[n * 3 + 0];
      bm.y = W_ih[n * 3 + 1];
    } else {
      bm.x = W_ih[n * 3 + 2];
      bm.y = b_ih[n];  // bias folded via A's K=3 == 1.0
    }
  } else {
    bm.x = 0.0f;
    bm.y = 0.0f;
  }

  // A fragment: row m = n, A[m][k] = inputs(b, s, k) for k<3, A[m][3] = 1.0
  const unsigned r = (unsigned)wave * 16u + (unsigned)n;
  const unsigned b = r / (unsigned)sc;
  const unsigned sl = r - b * (unsigned)sc;
  const float* ip = in + ((size_t)b * NS + (size_t)s0 + sl) * NI;
  v2f am;
  if (kh == 0) {
    am.x = ip[0];
    am.y = ip[1];
  } else {
    am.x = ip[2];
    am.y = 1.0f;
  }

  v8f acc = {};
  acc = __builtin_amdgcn_wmma_f32_16x16x4_f32(false, am, false, bm, (short)0,
                                              acc, false, false);

  // D layout: VGPR v -> row (v + 8*kh), col = n ; store cols 0..11
  const size_t base = (size_t)wave * 16u;
#pragma unroll
  for (int v = 0; v < 8; ++v) {
    if (n < NG) {
      const size_t row = base + (unsigned)(v + 8 * kh);
      xg[row * NG + n] = acc[v];
    }
  }
}

// ---------------------------------------------------------------------------
// Pass 2: GRU recurrence over one S-chunk. Quad (4 lanes) per batch element;
// lane j owns h_j. 512 single-wave blocks -> spread across WGPs.
// ---------------------------------------------------------------------------
__global__ __launch_bounds__(32) void tinygru_recurrence(
    const float* __restrict__ xg,    // [NB*sc, 12] chunk slab
    const float* __restrict__ W_hh,  // [12,4]
    const float* __restrict__ b_hh,  // [12]
    const float* __restrict__ W_ro,  // [2,4]
    const float* __restrict__ b_ro,  // [2]
    const float* __restrict__ h0p,   // [1,4]
    float* __restrict__ out,         // [B,S,2]
    float* __restrict__ hfin,        // [B,4]
    float* __restrict__ hcarry,      // [B,4] workspace carry
    int s0, int sc) {
  const int j = threadIdx.x & 3;                       // hidden index owned
  const int b = blockIdx.x * 8 + (threadIdx.x >> 2);   // batch element

  // Per-lane recurrent weights (rows j, 4+j, 8+j of W_hh) + biases.
  const float wr0 = W_hh[j * 4 + 0], wr1 = W_hh[j * 4 + 1],
              wr2 = W_hh[j * 4 + 2], wr3 = W_hh[j * 4 + 3];
  const float wz0 = W_hh[(4 + j) * 4 + 0], wz1 = W_hh[(4 + j) * 4 + 1],
              wz2 = W_hh[(4 + j) * 4 + 2], wz3 = W_hh[(4 + j) * 4 + 3];
  const float wn0 = W_hh[(8 + j) * 4 + 0], wn1 = W_hh[(8 + j) * 4 + 1],
              wn2 = W_hh[(8 + j) * 4 + 2], wn3 = W_hh[(8 + j) * 4 + 3];
  const float bhr = b_hh[j], bhz = b_hh[4 + j], bhn = b_hh[8 + j];

  // Readout: lane0 -> out[0], lane1 -> out[1] (lanes 2,3 compute dummy row 0)
  const int wsel = (j < 2) ? j : 0;
  const float wo0 = W_ro[wsel * 4 + 0], wo1 = W_ro[wsel * 4 + 1],
              wo2 = W_ro[wsel * 4 + 2], wo3 = W_ro[wsel * 4 + 3];
  const float bro = b_ro[wsel];

  float hq, h0v, h1v, h2v, h3v;
  if (s0 == 0) {
    hq = h0p[j];
    h0v = h0p[0]; h1v = h0p[1]; h2v = h0p[2]; h3v = h0p[3];
  } else {
    const float* hc = hcarry + b * NH;
    hq = hc[j];
    h0v = hc[0]; h1v = hc[1]; h2v = hc[2]; h3v = hc[3];
  }

  const float* xp = xg + (size_t)b * sc * NG;
  float* op = out + ((size_t)b * NS + s0) * NO;

  // software-pipeline the x_gates loads by one step
  float xr = xp[j], xz = xp[4 + j], xn = xp[8 + j];

  for (int s = 0; s < sc; ++s) {
    __builtin_prefetch(xp + 16 * NG, 0, 0);  // ~768B ahead in this quad stream
    float nxr = 0.0f, nxz = 0.0f, nxn = 0.0f;
    if (s + 1 < sc) {  // uniform branch
      nxr = xp[NG + j];
      nxz = xp[NG + 4 + j];
      nxn = xp[NG + 8 + j];
    }

    const float hr = __builtin_fmaf(wr3, h3v, __builtin_fmaf(wr2, h2v,
                     __builtin_fmaf(wr1, h1v, __builtin_fmaf(wr0, h0v, bhr))));
    const float hz = __builtin_fmaf(wz3, h3v, __builtin_fmaf(wz2, h2v,
                     __builtin_fmaf(wz1, h1v, __builtin_fmaf(wz0, h0v, bhz))));
    const float hn = __builtin_fmaf(wn3, h3v, __builtin_fmaf(wn2, h2v,
                     __builtin_fmaf(wn1, h1v, __builtin_fmaf(wn0, h0v, bhn))));

    const float rg = fsigmoid(xr + hr);
    const float zg = fsigmoid(xz + hz);
    const float ng = ftanh_fast(__builtin_fmaf(rg, hn, xn));
    hq = __builtin_fmaf(zg, hq - ng, ng);  // (1-z)*n + z*h

    // share h_new within the quad (DPP quad_perm broadcasts)
    h0v = qbcast<0>(hq);
    h1v = qbcast<1>(hq);
    h2v = qbcast<2>(hq);
    h3v = qbcast<3>(hq);

    // fused readout
    const float o = __builtin_fmaf(wo3, h3v, __builtin_fmaf(wo2, h2v,
                    __builtin_fmaf(wo1, h1v, __builtin_fmaf(wo0, h0v, bro))));
    if (j < NO) op[j] = o;

    xr = nxr; xz = nxz; xn = nxn;
    xp += NG;
    op += NO;
  }

  hcarry[b * NH + j] = hq;
  if (s0 + sc == NS) hfin[b * NH + j] = hq;
}

// ---------------------------------------------------------------------------
extern "C" void kernel_launch(void* const* d_in, const int* in_sizes, int n_in,
                              void* d_out, int out_size, void* d_ws,
                              size_t ws_size, hipStream_t stream) {
  (void)in_sizes; (void)n_in; (void)out_size;

  const float* in   = (const float*)d_in[0];
  const float* W_ih = (const float*)d_in[1];
  const float* W_hh = (const float*)d_in[2];
  const float* b_ih = (const float*)d_in[3];
  const float* b_hh = (const float*)d_in[4];
  const float* W_ro = (const float*)d_in[5];
  const float* b_ro = (const float*)d_in[6];
  const float* h0p  = (const float*)d_in[7];

  float* out    = (float*)d_out;
  float* hfin   = out + (size_t)NB * NS * NO;       // outputs then h_final
  float* hcarry = (float*)d_ws;                     // 64KB slab for h carry
  float* xg     = (float*)((char*)d_ws + 65536);    // x_gates chunk slab

  // chunk length: fit workspace, cap at 256 steps (50MB -> L2-resident)
  const size_t avail = (ws_size > 65536) ? (ws_size - 65536) : 0;
  long maxsc = (long)(avail / ((size_t)NB * NG * sizeof(float)));
  if (maxsc < 1) maxsc = 1;
  if (maxsc > 256) maxsc = 256;
  const int sc_max = (int)maxsc;

  for (int s0 = 0; s0 < NS; s0 += sc_max) {
    const int sc = (NS - s0 < sc_max) ? (NS - s0) : sc_max;
    const int tiles = (NB * sc) / 16;  // exact: NB multiple of 16
    tinygru_xgates_wmma<<<dim3(tiles / 8), 256, 0, stream>>>(
        in, W_ih, b_ih, xg, s0, sc);
    tinygru_recurrence<<<dim3(NB / 8), 32, 0, stream>>>(
        xg, W_hh, b_hh, W_ro, b_ro, h0p, out, hfin, hcarry, s0, sc);
  }
}